// GCN_26809185861879
// MI455X (gfx1250) — compile-verified
//
#include <hip/hip_runtime.h>

// GCN 2-layer forward for MI455X (gfx1250, wave32).
// Layer-2 GEMM (100000x64 @ 64x128) uses V_WMMA_F32_16X16X4_F32 (exact fp32).
// Edge aggregation uses global_atomic_add_f32; working set is L2-resident (192MB).

#define N_NODES   100000
#define F1        64
#define F2        128
#define NEG_SLOPE 0.01f

typedef __attribute__((ext_vector_type(2))) float v2f;
typedef __attribute__((ext_vector_type(8))) float v8f;

__device__ __forceinline__ float leaky(float v) {
    return v >= 0.0f ? v : v * NEG_SLOPE;
}

// ---------------- degree / norm ----------------

__global__ void k_deg_init(float* __restrict__ deg) {
    int i = blockIdx.x * blockDim.x + threadIdx.x;
    if (i < N_NODES) deg[i] = 1.0f;          // self-loop contributes 1
}

__global__ void k_deg_accum(const int* __restrict__ dst, int E,
                            float* __restrict__ deg) {
    int e = blockIdx.x * blockDim.x + threadIdx.x;
    if (e < E) atomicAdd(&deg[dst[e]], 1.0f);
}

__global__ void k_dinv(const float* __restrict__ deg, float* __restrict__ dinv) {
    int i = blockIdx.x * blockDim.x + threadIdx.x;
    if (i < N_NODES) dinv[i] = rsqrtf(fmaxf(deg[i], 1.0f));
}

// ---------------- layer 1: h1 = x @ W1 ; agg1 init with self-loop ----------------

__global__ void k_layer1(const float* __restrict__ x, const float* __restrict__ W1,
                         const float* __restrict__ dinv,
                         float* __restrict__ h1, float* __restrict__ agg1) {
    int idx = blockIdx.x * blockDim.x + threadIdx.x;
    if (idx >= N_NODES * F1) return;
    int n = idx >> 6;          // node
    int f = idx & (F1 - 1);    // feature
    float h = fmaf(x[2 * n], W1[f], x[2 * n + 1] * W1[F1 + f]);
    h1[idx] = h;
    float di = dinv[n];
    agg1[idx] = h * di * di;   // self-loop message
}

// ---------------- edge aggregation, layer 1 (64 feats/edge) ----------------

__global__ void k_edge_agg1(const int* __restrict__ src, const int* __restrict__ dst,
                            int E, const float* __restrict__ dinv,
                            const float* __restrict__ h1, float* __restrict__ agg1) {
    int t = blockIdx.x * 256 + threadIdx.x;
    int e = t >> 6;
    int f = t & (F1 - 1);
    if (e >= E) return;
    int s = src[e];
    int d = dst[e];
    float norm = dinv[s] * dinv[d];
    atomicAdd(&agg1[d * F1 + f], h1[s * F1 + f] * norm);
}

// ---------------- bias + leaky_relu, layer 1 (in place) ----------------

__global__ void k_act1(float* __restrict__ agg1, const float* __restrict__ b1) {
    int idx = blockIdx.x * blockDim.x + threadIdx.x;
    if (idx >= N_NODES * F1) return;
    agg1[idx] = leaky(agg1[idx] + b1[idx & (F1 - 1)]);
}

// ---------------- layer 2 GEMM via WMMA fp32: h2 = A(Nx64) @ W2(64x128) ----------------
// One wave computes a 16x16 output tile. Block = 256 threads = 8 waves = the 8
// column tiles of one 16-row stripe. N_NODES % 16 == 0 -> EXEC all ones (WMMA req).
// Epilogue also seeds agg2 (= d_out) with the self-loop term h2 * dinv^2.

__global__ void __launch_bounds__(256)
k_gemm2_wmma(const float* __restrict__ A,     // [N, 64]  (activated layer-1)
             const float* __restrict__ W2,    // [64, 128]
             const float* __restrict__ dinv,
             float* __restrict__ h2,          // [N, 128] raw GEMM result
             float* __restrict__ agg2) {      // [N, 128] accumulator (d_out)
    const int lane = threadIdx.x & 31;
    const int wave = threadIdx.x >> 5;        // 0..7 -> column tile
    const int row0 = blockIdx.x * 16;
    const int col0 = wave * 16;

    const int m  = lane & 15;                 // A row within tile
    const int kh = (lane >> 4) << 1;          // K sub-offset: 0 (lanes 0-15) / 2 (16-31)

    v8f acc = {};
    const float* Arow = A + (size_t)(row0 + m) * F1;
#pragma unroll
    for (int k0 = 0; k0 < F1; k0 += 4) {
        v2f a, b;
        a.x = Arow[k0 + kh];
        a.y = Arow[k0 + kh + 1];
        b.x = W2[(k0 + kh) * F2 + col0 + m];
        b.y = W2[(k0 + kh + 1) * F2 + col0 + m];
        acc = __builtin_amdgcn_wmma_f32_16x16x4_f32(
            /*neg_a=*/false, a, /*neg_b=*/false, b,
            /*c_mod=*/(short)0, acc, /*reuse_a=*/false, /*reuse_b=*/false);
    }

    // C/D layout: VGPR v holds row M = v + 8*(lane>=16), col N = lane&15
    const int nc    = lane & 15;
    const int rbase = row0 + ((lane >> 4) << 3);
#pragma unroll
    for (int v = 0; v < 8; ++v) {
        int   r  = rbase + v;
        float d  = acc[v];
        float di = dinv[r];
        size_t o = (size_t)r * F2 + col0 + nc;
        h2[o]    = d;
        agg2[o]  = d * di * di;               // self-loop message
    }
}

// ---------------- edge aggregation, layer 2 (128 feats/edge) ----------------

__global__ void k_edge_agg2(const int* __restrict__ src, const int* __restrict__ dst,
                            int E, const float* __restrict__ dinv,
                            const float* __restrict__ h2, float* __restrict__ agg2) {
    int t = blockIdx.x * 256 + threadIdx.x;
    int e = t >> 7;
    int f = t & (F2 - 1);
    if (e >= E) return;
    int s = src[e];
    int d = dst[e];
    float norm = dinv[s] * dinv[d];
    atomicAdd(&agg2[d * F2 + f], h2[s * F2 + f] * norm);
}

// ---------------- final bias + leaky_relu (in place on d_out) ----------------

__global__ void k_act2(float* __restrict__ z, const float* __restrict__ b2) {
    int idx = blockIdx.x * blockDim.x + threadIdx.x;
    if (idx >= N_NODES * F2) return;
    z[idx] = leaky(z[idx] + b2[idx & (F2 - 1)]);
}

// ---------------- host launcher ----------------

extern "C" void kernel_launch(void* const* d_in, const int* in_sizes, int n_in,
                              void* d_out, int out_size, void* d_ws, size_t ws_size,
                              hipStream_t stream) {
    const int*   edge_index = (const int*)d_in[0];   // [2, E] row-major (int32)
    const float* x          = (const float*)d_in[1]; // [N, 2]
    const float* W1         = (const float*)d_in[2]; // [2, 64]
    const float* b1         = (const float*)d_in[3]; // [64]
    const float* W2         = (const float*)d_in[4]; // [64, 128]
    const float* b2         = (const float*)d_in[5]; // [128]
    float*       z          = (float*)d_out;         // [N, 128]

    const int E = in_sizes[0] / 2;
    const int* src = edge_index;
    const int* dst = edge_index + E;

    // Workspace layout (floats). h2 overlaps h1 (h1 dead before GEMM writes h2).
    float* ws   = (float*)d_ws;
    float* deg  = ws;                         // [N]
    float* dinv = ws + N_NODES;               // [N]
    float* agg1 = ws + 2 * N_NODES;           // [N*64]   (becomes activated h)
    float* h1   = agg1 + N_NODES * F1;        // [N*64]
    float* h2   = h1;                         // [N*128]  reuses h1 region + beyond
    (void)ws_size; (void)n_in; (void)out_size;

    const int T = 256;

    // degree + symmetric norm
    k_deg_init <<<(N_NODES + T - 1) / T, T, 0, stream>>>(deg);
    k_deg_accum<<<(E + T - 1) / T, T, 0, stream>>>(dst, E, deg);
    k_dinv     <<<(N_NODES + T - 1) / T, T, 0, stream>>>(deg, dinv);

    // layer 1
    k_layer1   <<<(N_NODES * F1 + T - 1) / T, T, 0, stream>>>(x, W1, dinv, h1, agg1);
    k_edge_agg1<<<((size_t)E * F1 + T - 1) / T, T, 0, stream>>>(src, dst, E, dinv, h1, agg1);
    k_act1     <<<(N_NODES * F1 + T - 1) / T, T, 0, stream>>>(agg1, b1);

    // layer 2: WMMA GEMM (6250 row-stripes x 8 col-tiles), then edge aggregation
    k_gemm2_wmma<<<N_NODES / 16, 256, 0, stream>>>(agg1, W2, dinv, h2, z);
    k_edge_agg2 <<<((size_t)E * F2 + T - 1) / T, T, 0, stream>>>(src, dst, E, dinv, h2, z);
    k_act2      <<<(N_NODES * F2 + T - 1) / T, T, 0, stream>>>(z, b2);
}